// SMap3x3_34333968564896
// MI455X (gfx1250) — compile-verified
//
#include <hip/hip_runtime.h>
#include <stdint.h>

// SMap3x3 fused two-phase 3x3 stencil for MI455X (gfx1250, wave32).
// Bandwidth-bound (~38MB total traffic -> ~1.6us HBM roofline); no matrix
// structure, so the CDNA5 path exercised is async global->LDS staging
// (global_load_async_to_lds_b32 + s_wait_asynccnt) instead of WMMA.

namespace {

constexpr int HIN = 382, WIN = 382;   // unpadded input
constexpr int HP  = 384, WP  = 384;   // padded grid
constexpr int NIMG = 8;               // B*C
constexpr int TILE = 32;
constexpr int AH = TILE + 2;          // 34: phase-A region (halo 1)
constexpr int ZH = TILE + 4;          // 36: z region (halo 2)
constexpr int NTHREADS = 256;         // 8 waves of 32

__device__ __forceinline__ uint32_t lds_byte_off(const void* p) {
  // generic pointer to LDS: low 32 bits are the AS3 byte offset
  return (uint32_t)(uintptr_t)p;
}

__device__ __forceinline__ void async_copy_b32(uint32_t lds_addr, const float* g) {
#if defined(__HIP_DEVICE_COMPILE__)
  asm volatile("global_load_async_to_lds_b32 %0, %1, off"
               :: "v"(lds_addr), "v"((uint64_t)(uintptr_t)g)
               : "memory");
#else
  (void)lds_addr; (void)g;
#endif
}

__device__ __forceinline__ void async_wait0() {
#if defined(__HIP_DEVICE_COMPILE__)
  asm volatile("s_wait_asynccnt 0x0" ::: "memory");
#endif
}

__global__ __launch_bounds__(NTHREADS)
void smap3x3_fused(const float* __restrict__ xin, const float* __restrict__ yin,
                   const float* __restrict__ zin, const float* __restrict__ rin,
                   const float* __restrict__ kinv, float* __restrict__ out)
{
  __shared__ float z_s[ZH * ZH];
  __shared__ float X_s[ZH * ZH];
  __shared__ float Y_s[ZH * ZH];
  __shared__ float x_s[AH * AH];
  __shared__ float y_s[AH * AH];
  __shared__ float rm_s[AH * AH];
  __shared__ int   best_s[AH * AH];

  const int img = blockIdx.z;
  const int h0  = blockIdx.y * TILE;  // tile origin in padded coords
  const int w0  = blockIdx.x * TILE;
  const int tid = threadIdx.x;

  const float k00 = kinv[0], k01 = kinv[1], k02 = kinv[2];
  const float k10 = kinv[3], k11 = kinv[4], k12 = kinv[5];

  const size_t base = (size_t)img * (HIN * WIN);
  const float* __restrict__ xp = xin + base;
  const float* __restrict__ yp = yin + base;
  const float* __restrict__ zp = zin + base;
  const float* __restrict__ rp = rin + base;

  // Interior test: all staged coords (incl. halo-2) fall inside the original
  // 382x382 image, so async bulk copies need no zero-fill.
  const bool interior =
      (h0 >= 3) && (w0 >= 3) && (h0 + ZH - 3 <= HIN) && (w0 + ZH - 3 <= WIN);

  if (interior) {
    for (int idx = tid; idx < ZH * ZH; idx += NTHREADS) {
      const int ly = idx / ZH, lx = idx % ZH;
      async_copy_b32(lds_byte_off(&z_s[idx]),
                     &zp[(h0 - 3 + ly) * WIN + (w0 - 3 + lx)]);
    }
    for (int idx = tid; idx < AH * AH; idx += NTHREADS) {
      const int ly = idx / AH, lx = idx % AH;
      const int o = (h0 - 2 + ly) * WIN + (w0 - 2 + lx);
      async_copy_b32(lds_byte_off(&x_s[idx]),  &xp[o]);
      async_copy_b32(lds_byte_off(&y_s[idx]),  &yp[o]);
      async_copy_b32(lds_byte_off(&rm_s[idx]), &rp[o]);
    }
    async_wait0();
  } else {
    for (int idx = tid; idx < ZH * ZH; idx += NTHREADS) {
      const int ly = idx / ZH, lx = idx % ZH;
      const int gh = h0 - 2 + ly, gw = w0 - 2 + lx;
      float v = 0.0f;
      if (gh >= 1 && gh <= HIN && gw >= 1 && gw <= WIN)
        v = zp[(gh - 1) * WIN + (gw - 1)];
      z_s[idx] = v;
    }
    for (int idx = tid; idx < AH * AH; idx += NTHREADS) {
      const int ly = idx / AH, lx = idx % AH;
      const int gh = h0 - 1 + ly, gw = w0 - 1 + lx;
      float xv = 0.0f, yv = 0.0f, rv = 0.0f;
      if (gh >= 1 && gh <= HIN && gw >= 1 && gw <= WIN) {
        const int o = (gh - 1) * WIN + (gw - 1);
        xv = xp[o]; yv = yp[o]; rv = rp[o];
      }
      x_s[idx] = xv; y_s[idx] = yv; rm_s[idx] = rv;
    }
  }
  __syncthreads();

  // Projected X,Y per staged pixel (z=0 outside grid -> X=Y=0 == shift fill).
  for (int idx = tid; idx < ZH * ZH; idx += NTHREADS) {
    const int ly = idx / ZH, lx = idx % ZH;
    const float u = (float)(w0 - 2 + lx) - 0.5f;
    const float v = (float)(h0 - 2 + ly) - 0.5f;
    const float zq = z_s[idx];
    X_s[idx] = (k00 * u + k01 * v + k02) * zq;
    Y_s[idx] = (k10 * u + k11 * v + k12) * zq;
  }
  __syncthreads();

  // Phase A: best-of-9 per pixel in the 34x34 halo-1 region.
  for (int idx = tid; idx < AH * AH; idx += NTHREADS) {
    const int ay = idx / AH, ax = idx % AH;
    const float xq = x_s[idx], yq = y_s[idx];
    const int zb = (ay + 1) * ZH + (ax + 1);
    int best = 0;
    float bv = 3.402823466e38f;
#pragma unroll
    for (int k = 0; k < 9; ++k) {
      const int di = 1 - k / 3, dj = 1 - k % 3;
      const int n = zb + di * ZH + dj;
      const float key = fabsf(xq - X_s[n]) + fabsf(yq - Y_s[n]);
      if (key < bv) { bv = key; best = k; }   // first-min tie-break
    }
    best_s[idx] = best;
  }
  __syncthreads();

  // Phase B: per output pixel, scatter-as-gather over the 9 neighbors.
  for (int pidx = tid; pidx < TILE * TILE; pidx += NTHREADS) {
    const int ty = pidx / TILE, tx = pidx % TILE;
    const int ph = h0 + ty, pw = w0 + tx;

    int   idxk   = 0;
    float bz     = 3.402823466e38f;
    float sum_rm = 0.0f;
#pragma unroll
    for (int k = 0; k < 9; ++k) {
      const int dh = 1 - k / 3, dw = 1 - k % 3;
      const int qh = ph + dh, qw = pw + dw;
      float az = 1.0e7f, arm = 0.0f;
      if (qh >= 0 && qh < HP && qw >= 0 && qw < WP) {
        const int qa  = (ty + 1 + dh) * AH + (tx + 1 + dw);
        const float rmq = rm_s[qa];
        if (rmq > 0.5f) {
          if (best_s[qa] == k) {
            az  = z_s[(ty + 2 + dh) * ZH + (tx + 2 + dw)];
            arm = rmq;
          }
        } else if (k == 4) {
          arm = rmq;   // forced-center channel: new_rm_4 = rm
        }
      }
      sum_rm += arm;
      if (az < bz) { bz = az; idxk = k; }      // first-min tie-break
    }
    if (!(sum_rm > 0.5f)) idxk = 4;

    // Gather the 4 output channels from the selected neighbor.
    const int dh = 1 - idxk / 3, dw = 1 - idxk % 3;
    const int qh = ph + dh, qw = pw + dw;
    float o0 = 0.0f, o1 = 0.0f, o2 = 0.0f, o3 = 0.0f;
    if (qh >= 0 && qh < HP && qw >= 0 && qw < WP) {
      const int qa  = (ty + 1 + dh) * AH + (tx + 1 + dw);
      const float rmq = rm_s[qa];
      const int   bq  = best_s[qa];
      if (rmq > 0.5f) {
        if (bq == idxk) {
          o0 = x_s[qa];
          o1 = y_s[qa];
          o2 = z_s[(ty + 2 + dh) * ZH + (tx + 2 + dw)];
          o3 = rmq;
        }
      } else if (idxk == 4) {
        o3 = rmq;
      }
    }

    const size_t plane = (size_t)HP * WP;
    const size_t ob = (size_t)img * 4 * plane + (size_t)ph * WP + pw;
    out[ob]             = o0;
    out[ob + plane]     = o1;
    out[ob + 2 * plane] = o2;
    out[ob + 3 * plane] = o3;
  }
}

} // namespace

extern "C" void kernel_launch(void* const* d_in, const int* in_sizes, int n_in,
                              void* d_out, int out_size, void* d_ws, size_t ws_size,
                              hipStream_t stream) {
  (void)in_sizes; (void)n_in; (void)out_size; (void)d_ws; (void)ws_size;
  const float* x  = (const float*)d_in[0];
  const float* y  = (const float*)d_in[1];
  const float* z  = (const float*)d_in[2];
  const float* r  = (const float*)d_in[3];
  const float* ki = (const float*)d_in[4];
  float* out = (float*)d_out;

  dim3 grid(WP / TILE, HP / TILE, NIMG);  // 12 x 12 x 8
  hipLaunchKernelGGL(smap3x3_fused, grid, dim3(NTHREADS), 0, stream,
                     x, y, z, r, ki, out);
}